// NodesToEdges_57329223467232
// MI455X (gfx1250) — compile-verified
//
#include <hip/hip_runtime.h>
#include <hip/hip_bf16.h>

typedef __attribute__((ext_vector_type(2))) float v2f;
typedef __attribute__((ext_vector_type(8))) float v8f;

// Rows r = e*3 + d of the (3M x 32) "A" matrix.
// Per wave: one 16-row tile per grid-stride iteration.
// D(16x32) = A_src(16x32) @ P(32x32) + A_dst(16x32) @ Q(32x32)
// where A_src = W ⊙ xn[src], A_dst = W ⊙ xn[dst],
//       P = 0.5*M1 + 0.25*M2, Q = -0.5*M1 + 0.25*M2.
__global__ __launch_bounds__(256) void NodesToEdges_wmma_kernel(
    const float* __restrict__ xn,      // (N_NODES, 3, 32)
    const int*   __restrict__ xe_src,  // (N_EDGES)
    const int*   __restrict__ xe_dst,  // (N_EDGES)
    const float* __restrict__ W,       // (N_EDGES, 32)
    const float* __restrict__ M1,      // (32, 32)
    const float* __restrict__ M2,      // (32, 32)
    float*       __restrict__ out,     // (N_EDGES, 3, 32)
    unsigned totalRows, unsigned numTiles)
{
  const int lane   = threadIdx.x & 31;
  const int laneLo = lane & 15;
  const int laneHi = lane >> 4;          // 0: K%4 in {0,1}; 1: K%4 in {2,3}
  const int colA   = laneHi * 2;

  const unsigned wavesPerBlock = blockDim.x >> 5;
  const unsigned waveId   = blockIdx.x * wavesPerBlock + (threadIdx.x >> 5);
  const unsigned numWaves = gridDim.x * wavesPerBlock;

  // ---- Build B operands (P and Q) in registers, once per wave. ----
  // B layout for 32-bit 4x16: VGPR0 lane n holds B[K = 4s + 2*laneHi][N = 16t + laneLo],
  //                           VGPR1 holds K+1.  (v2f = {VGPR0, VGPR1} per lane)
  v2f bP[2][8], bQ[2][8];
#pragma unroll
  for (int t = 0; t < 2; ++t) {
#pragma unroll
    for (int s = 0; s < 8; ++s) {
      const int k0  = 4 * s + colA;
      const int col = t * 16 + laneLo;
      const float m1a = M1[k0 * 32 + col];
      const float m1b = M1[(k0 + 1) * 32 + col];
      const float m2a = M2[k0 * 32 + col];
      const float m2b = M2[(k0 + 1) * 32 + col];
      bP[t][s].x =  0.5f * m1a + 0.25f * m2a;
      bP[t][s].y =  0.5f * m1b + 0.25f * m2b;
      bQ[t][s].x = -0.5f * m1a + 0.25f * m2a;
      bQ[t][s].y = -0.5f * m1b + 0.25f * m2b;
    }
  }

  for (unsigned tile = waveId; tile < numTiles; tile += numWaves) {
    const unsigned rowBase = tile * 16u;
    unsigned myRow = rowBase + (unsigned)laneLo;
    if (myRow >= totalRows) myRow = totalRows - 1u;  // clamp loads; stores guarded
    const unsigned e = myRow / 3u;                   // 32-bit magic-mul division
    const unsigned d = myRow - e * 3u;

    const unsigned si = (unsigned)xe_src[e];
    const unsigned di = (unsigned)xe_dst[e];
    const float* __restrict__ srcRow = xn + ((size_t)si * 3u + d) * 32u;
    const float* __restrict__ dstRow = xn + ((size_t)di * 3u + d) * 32u;
    const float* __restrict__ wRow   = W + (size_t)e * 32u;

    // ---- Gather A rows: per lane, 8 x (b64 from W, xn[src], xn[dst]) ----
    v2f aS[8], aD[8];
#pragma unroll
    for (int s = 0; s < 8; ++s) {
      const int k = 4 * s + colA;
      const v2f w2 = *(const v2f*)(wRow   + k);
      const v2f xs = *(const v2f*)(srcRow + k);
      const v2f xd = *(const v2f*)(dstRow + k);
      aS[s] = w2 * xs;
      aD[s] = w2 * xd;
    }

    // ---- 32 x V_WMMA_F32_16X16X4_F32: two 16-col output tiles, K=32 ----
    v8f acc0 = {}, acc1 = {};
#pragma unroll
    for (int s = 0; s < 8; ++s) {
      acc0 = __builtin_amdgcn_wmma_f32_16x16x4_f32(false, aS[s], false, bP[0][s],
                                                   (short)0, acc0, false, false);
      acc1 = __builtin_amdgcn_wmma_f32_16x16x4_f32(false, aS[s], false, bP[1][s],
                                                   (short)0, acc1, false, false);
      acc0 = __builtin_amdgcn_wmma_f32_16x16x4_f32(false, aD[s], false, bQ[0][s],
                                                   (short)0, acc0, false, false);
      acc1 = __builtin_amdgcn_wmma_f32_16x16x4_f32(false, aD[s], false, bQ[1][s],
                                                   (short)0, acc1, false, false);
    }

    // ---- Store D: VGPR v, lane -> D[M = v + 8*laneHi][N = laneLo (+16)] ----
    // Output is written once and never re-read: use non-temporal stores so the
    // 384 MB output stream doesn't evict the L2-resident node table.
    const unsigned rBase = rowBase + 8u * (unsigned)laneHi;
    if (rowBase + 16u <= totalRows) {
      // fast path: full tile, no guards
#pragma unroll
      for (int v = 0; v < 8; ++v) {
        float* p = out + (size_t)(rBase + v) * 32u + laneLo;
        __builtin_nontemporal_store(acc0[v], p);
        __builtin_nontemporal_store(acc1[v], p + 16);
      }
    } else {
#pragma unroll
      for (int v = 0; v < 8; ++v) {
        if (rBase + v < totalRows) {
          float* p = out + (size_t)(rBase + v) * 32u + laneLo;
          __builtin_nontemporal_store(acc0[v], p);
          __builtin_nontemporal_store(acc1[v], p + 16);
        }
      }
    }
  }
}

extern "C" void kernel_launch(void* const* d_in, const int* in_sizes, int n_in,
                              void* d_out, int out_size, void* d_ws, size_t ws_size,
                              hipStream_t stream) {
  const float* xn     = (const float*)d_in[0];
  const int*   xe_src = (const int*)d_in[1];
  const int*   xe_dst = (const int*)d_in[2];
  const float* W      = (const float*)d_in[3];
  const float* M1     = (const float*)d_in[4];
  const float* M2     = (const float*)d_in[5];
  float*       out    = (float*)d_out;

  const unsigned nEdges    = (unsigned)(in_sizes[3] / 32);  // W is (N_EDGES, 32)
  const unsigned totalRows = nEdges * 3u;
  const unsigned numTiles  = (totalRows + 15u) / 16u;

  const int threads = 256;                        // 8 waves / block (wave32)
  unsigned blocks = (numTiles + 7u) / 8u;         // >= 1 tile per wave
  if (blocks > 4096u) blocks = 4096u;             // grid-stride amortizes B-reg setup
  if (blocks < 1u) blocks = 1u;

  NodesToEdges_wmma_kernel<<<blocks, threads, 0, stream>>>(
      xn, xe_src, xe_dst, W, M1, M2, out, totalRows, numTiles);
}